// GCN_69767448756417
// MI455X (gfx1250) — compile-verified
//
#include <hip/hip_runtime.h>
#include <cstddef>

// ---------------------------------------------------------------------------
// GCN 2-layer forward for gfx1250 (MI455X).
//   h1  = x @ W1                      (V_WMMA_F32_16X16X4_F32, fully unrolled K)
//   a1  = scatter_add(h1[row]*norm)   (global_atomic_add_f32, L2-resident)
//   r1  = relu(a1 + h1*dinv^2 + b1)
//   h2  = r1 @ W2                     (V_WMMA_F32_16X16X4_F32, fully unrolled K)
//   out = scatter_add(h2[row]*norm) + h2*dinv^2 + b2
// ---------------------------------------------------------------------------

typedef __attribute__((ext_vector_type(2))) float v2f;
typedef __attribute__((ext_vector_type(8))) float v8f;

__global__ void k_init_deg(float* __restrict__ deg, int n) {
  int i = blockIdx.x * blockDim.x + threadIdx.x;
  if (i < n) deg[i] = 1.0f;  // self-loop contributes 1 to every node's degree
}

__global__ void k_count_deg(const int* __restrict__ cols, float* __restrict__ deg, int E) {
  int e = blockIdx.x * blockDim.x + threadIdx.x;
  if (e < E) atomicAdd(&deg[cols[e]], 1.0f);
}

__global__ void k_rsqrt_inplace(float* __restrict__ deg, int n) {
  int i = blockIdx.x * blockDim.x + threadIdx.x;
  if (i < n) deg[i] = 1.0f / sqrtf(deg[i]);  // deg >= 1 always (self-loops)
}

// C[M,NN] = A[M,K] @ B[K,NN] with fp32 WMMA, K and NN compile-time so the
// K-loop fully unrolls into straight-line v_wmma chains with immediate-offset
// loads (no scalar loop control, no runtime-K fallback path).
// grid.x = M/16 tiles; blockDim.x = 32*(NN/16); wave w computes N-tile w.
// A frag (16x4): lane holds M=lane%16, K = kh..kh+1 with kh = 2*(lane/16).
// B frag (4x16): lane holds N=lane%16, same K split.
// C/D (16x16, 8 VGPRs): M = v + 8*(lane/16), N = lane%16.
template <int K, int NN>
__global__ void k_gemm_wmma(const float* __restrict__ A, const float* __restrict__ B,
                            float* __restrict__ C) {
  const int wave = threadIdx.x >> 5;
  const int lane = threadIdx.x & 31;
  const int m0   = blockIdx.x << 4;
  const int n0   = wave << 4;
  const int l16  = lane & 15;
  const int kh   = (lane >> 4) << 1;  // 0 or 2

  v8f acc = {};
  const float* arow = A + (size_t)(m0 + l16) * K + kh;
  const float* bcol = B + (size_t)kh * NN + n0 + l16;

#pragma unroll
  for (int k = 0; k < K; k += 4) {
    v2f a, b;
    a.x = arow[k];
    a.y = arow[k + 1];
    b.x = bcol[(size_t)k * NN];
    b.y = bcol[(size_t)(k + 1) * NN];
    // 8 args: (neg_a, A, neg_b, B, c_mod, C, reuse_a, reuse_b)
    acc = __builtin_amdgcn_wmma_f32_16x16x4_f32(false, a, false, b, (short)0, acc,
                                                false, false);
  }

  const int chi = (lane >> 4) << 3;  // 0 or 8
  float* crow = C + (size_t)(m0 + chi) * NN + n0 + l16;
#pragma unroll
  for (int v = 0; v < 8; ++v) crow[(size_t)v * NN] = acc[v];
}

// One lane per (edge, feature). F = 1<<logF features per row.
__global__ void k_scatter(const int* __restrict__ rows, const int* __restrict__ cols,
                          const float* __restrict__ dinv, const float* __restrict__ H,
                          float* __restrict__ AGG, int E, int logF) {
  const int F   = 1 << logF;
  const int epb = blockDim.x >> logF;
  const int e   = blockIdx.x * epb + (threadIdx.x >> logF);
  if (e >= E) return;
  const int f = threadIdx.x & (F - 1);
  const int r = rows[e];
  const int c = cols[e];
  const float w = dinv[r] * dinv[c];
  atomicAdd(&AGG[(size_t)c * F + f], H[(size_t)r * F + f] * w);
}

// OUT[i] = (AGG[i] + H[i]*dinv^2 + bias[f])  (+ optional ReLU). In-place safe.
__global__ void k_post(const float* __restrict__ AGG, const float* __restrict__ H,
                       const float* __restrict__ dinv, const float* __restrict__ bias,
                       float* __restrict__ OUT, int n, int logF, int do_relu) {
  const int F    = 1 << logF;
  const size_t i = (size_t)blockIdx.x * blockDim.x + threadIdx.x;
  if (i >= ((size_t)n << logF)) return;
  const int node = (int)(i >> logF);
  const int f    = (int)(i & (size_t)(F - 1));
  const float d  = dinv[node];
  float v = AGG[i] + H[i] * d * d + bias[f];
  if (do_relu) v = fmaxf(v, 0.0f);
  OUT[i] = v;
}

extern "C" void kernel_launch(void* const* d_in, const int* in_sizes, int n_in,
                              void* d_out, int out_size, void* d_ws, size_t ws_size,
                              hipStream_t stream) {
  const float* x  = (const float*)d_in[0];
  const int*   ei = (const int*)d_in[1];   // int32 per harness convention
  const float* W1 = (const float*)d_in[2];
  const float* b1 = (const float*)d_in[3];
  const float* W2 = (const float*)d_in[4];
  const float* b2 = (const float*)d_in[5];

  const int IN_DIM = 128, HID = 64, OUT = 32;
  const int N = in_sizes[0] / IN_DIM;   // 50000, divisible by 16
  const int E = in_sizes[1] / 2;        // 800000
  const int* rows = ei;                 // edge_index[0] = source
  const int* cols = ei + E;             // edge_index[1] = target

  // Workspace layout (floats): [dinv: N][h1: N*64 -> reused as h2:N*32 | agg2:N*32][agg1: N*64]
  float* ws   = (float*)d_ws;
  float* dinv = ws;
  float* h1   = ws + N;
  float* agg1 = h1 + (size_t)N * HID;
  float* h2   = h1;                       // h1 dead after post1
  float* agg2 = h1 + (size_t)N * OUT;

  const int T = 256;

  // --- normalization: dinv = 1/sqrt(deg) with self-loops ---
  k_init_deg<<<(N + T - 1) / T, T, 0, stream>>>(dinv, N);
  k_count_deg<<<(E + T - 1) / T, T, 0, stream>>>(cols, dinv, E);
  k_rsqrt_inplace<<<(N + T - 1) / T, T, 0, stream>>>(dinv, N);

  // --- layer 1: GEMM (WMMA) -> scatter -> bias/self-loop/ReLU ---
  k_gemm_wmma<128, 64><<<N / 16, 32 * (64 / 16), 0, stream>>>(x, W1, h1);
  hipMemsetAsync(agg1, 0, (size_t)N * HID * sizeof(float), stream);
  k_scatter<<<(E * HID + T - 1) / T, T, 0, stream>>>(rows, cols, dinv, h1, agg1, E, 6);
  k_post<<<(int)(((size_t)N * HID + T - 1) / T), T, 0, stream>>>(agg1, h1, dinv, b1,
                                                                 agg1, N, 6, 1);

  // --- layer 2: GEMM (WMMA) -> scatter -> bias/self-loop ---
  k_gemm_wmma<64, 32><<<N / 16, 32 * (32 / 16), 0, stream>>>(agg1, W2, h2);
  hipMemsetAsync(agg2, 0, (size_t)N * OUT * sizeof(float), stream);
  k_scatter<<<(E * OUT + T - 1) / T, T, 0, stream>>>(rows, cols, dinv, h2, agg2, E, 5);
  k_post<<<(int)(((size_t)N * OUT + T - 1) / T), T, 0, stream>>>(agg2, h2, dinv, b2,
                                                                 (float*)d_out, N, 5, 0);
}